// CandidateFinder_33294586479004
// MI455X (gfx1250) — compile-verified
//
#include <hip/hip_runtime.h>
#include <stdint.h>

typedef float v2f __attribute__((ext_vector_type(2)));
typedef float v8f __attribute__((ext_vector_type(8)));

#define S     4096
#define NG    8
#define KMAX  32
#define TQ    16
#define NEG_INF (-__builtin_inff())

// ---- workspace byte offsets ----
#define OFF_QHASH   (0)                          // S*NG u32   = 128KB
#define OFF_KHASH   (OFF_QHASH + S*NG*4)         // S*NG u32   = 128KB
#define OFF_QSIGN   (OFF_KHASH + S*NG*4)         // S*NG u8    = 32KB
#define OFF_KSIGN   (OFF_QSIGN + S*NG)           // S u8       = 4KB
#define OFF_COLEQ   (OFF_KSIGN + S)              // 8 u32 (pad 64B)
#define OFF_ORD     (OFF_COLEQ + 64)             // 32 int (pad 128B)
#define OFF_CAND    (OFF_ORD + 128)              // S*256 int  = 4MB
#define OFF_SORTED  (OFF_CAND + S*256*4)         // S*256 int  = 4MB

// ---------------- kernel 1: hashes + sign bytes ----------------
__global__ void fa_hash_kernel(const float* __restrict__ q, const float* __restrict__ k,
                               const float* __restrict__ W,
                               uint32_t* __restrict__ qhash, uint32_t* __restrict__ khash,
                               uint8_t* __restrict__ qsign, uint8_t* __restrict__ ksign,
                               uint32_t* __restrict__ colEq) {
  const int gid   = blockIdx.x * blockDim.x + threadIdx.x;   // 0..S*NG-1
  const int isKey = blockIdx.y;
  if (gid == 0 && isKey == 0) {
    #pragma unroll
    for (int w = 0; w < 8; ++w) colEq[w] = 0xFFFFFFFFu;      // dup mask starts all-true
  }
  if (gid >= S * NG) return;
  const int i = gid >> 3, g = gid & 7;
  const float* x = (isKey ? k : q) + i * 64 + g * 8;
  float v[8];
  #pragma unroll
  for (int d = 0; d < 8; ++d) v[d] = x[d];

  uint32_t pack = 0;
  #pragma unroll
  for (int h = 0; h < 4; ++h) {
    float dot = 0.f;
    #pragma unroll
    for (int d = 0; d < 8; ++d) dot += v[d] * W[(g * 8 + d) * 4 + h];
    long long f = (long long)floorf(dot);
    int hv = (int)(f % 64);
    if (hv < 0) hv += 64;                                    // python-style mod
    pack |= ((uint32_t)hv) << (8 * h);
  }
  uint32_t sgn = 0;
  #pragma unroll
  for (int d = 0; d < 8; ++d) sgn |= (v[d] > 0.f ? 1u : 0u) << d;

  if (isKey) { khash[gid] = pack; if (g == 0) ksign[i] = (uint8_t)sgn; }
  else       { qhash[gid] = pack; qsign[gid] = (uint8_t)sgn; }
}

// ------------- kernel 2: WMMA similarity + masked top-32 per (row,group) -------------
__global__ void __launch_bounds__(256)
fa_cand_kernel(const float* __restrict__ q, const float* __restrict__ k,
               const uint32_t* __restrict__ qhash, const uint32_t* __restrict__ khash,
               const uint8_t* __restrict__ qsign, const uint8_t* __restrict__ ksign,
               int* __restrict__ cand) {
  const int g       = blockIdx.y;
  const int rowBase = blockIdx.x * TQ;
  const int tid     = threadIdx.x;
  const int wave    = tid >> 5;
  const int lane    = tid & 31;

  __shared__ float    qv[TQ][8];
  __shared__ uint32_t qh[TQ];
  __shared__ uint8_t  qs[TQ];
  __shared__ float    sim[TQ][128];      // 8 waves x 16 cols
  __shared__ float    topv[TQ][KMAX];
  __shared__ int      topi[TQ][KMAX];

  if (tid < TQ * 8) {
    int r = tid >> 3, d = tid & 7;
    qv[r][d] = q[(rowBase + r) * 64 + g * 8 + d];
  }
  if (tid < TQ) {
    qh[tid] = qhash[(rowBase + tid) * 8 + g];
    qs[tid] = qsign[(rowBase + tid) * 8 + g];
  }
  for (int t = tid; t < TQ * KMAX; t += 256) {
    topv[t >> 5][t & 31] = NEG_INF;
    topi[t >> 5][t & 31] = -1;
  }
  __syncthreads();

  // A fragments (16x4 f32 layout: lanes 0-15 K={0,1}, lanes 16-31 K={2,3})
  const int aM = lane & 15;
  const int aK = (lane >> 4) * 2;
  v2f a0; a0.x = qv[aM][aK];     a0.y = qv[aM][aK + 1];
  v2f a1; a1.x = qv[aM][aK + 4]; a1.y = qv[aM][aK + 5];

  const int N    = lane & 15;
  const int half = lane >> 4;

  for (int t = 0; t < 32; ++t) {
    const int ktile = t * 8 + wave;          // 0..255
    const int j     = ktile * 16 + N;        // this lane's key column
    const float* kp = k + j * 64 + g * 8;

    v2f b0; b0.x = kp[aK];     b0.y = kp[aK + 1];
    v2f b1; b1.x = kp[aK + 4]; b1.y = kp[aK + 5];

    if (t + 1 < 32) {                         // uniform condition: EXEC stays full
      const int jn = (t * 8 + wave + 8) * 16 + N;
      __builtin_prefetch(k + jn * 64 + g * 8, 0, 1);
    }

    v8f c = 0.0f;
    c = __builtin_amdgcn_wmma_f32_16x16x4_f32(false, a0, false, b0, (short)0, c, false, false);
    c = __builtin_amdgcn_wmma_f32_16x16x4_f32(false, a1, false, b1, (short)0, c, false, false);

    const uint32_t kh = khash[j * 8 + g];
    const uint8_t  ks = ksign[j];

    #pragma unroll
    for (int v = 0; v < 8; ++v) {
      const int r = v + 8 * half;                            // C-layout row
      const uint32_t x = qh[r] ^ kh;
      const bool lsh = ((x - 0x01010101u) & ~x & 0x80808080u) != 0;  // any hash byte equal
      const bool tri = (qs[r] == ks);                        // sign-pattern (trie) match
      sim[r][wave * 16 + N] = (lsh && tri) ? c[v] : NEG_INF;
    }
    __syncthreads();

    // 16 row-owners stream exact top-32 (strict > keeps lowest-index on ties, like top_k)
    if (tid < TQ) {
      const int r = tid;
      for (int s = 0; s < 128; ++s) {
        const float val = sim[r][s];
        if (val > topv[r][KMAX - 1]) {
          int pos = KMAX - 1;
          while (pos > 0 && val > topv[r][pos - 1]) {
            topv[r][pos] = topv[r][pos - 1];
            topi[r][pos] = topi[r][pos - 1];
            --pos;
          }
          topv[r][pos] = val;
          topi[r][pos] = t * 128 + s;        // == global key index
        }
      }
    }
    __syncthreads();
  }

  if (tid < TQ) {
    const int base = ((rowBase + tid) * NG + g) * KMAX;
    for (int s = 0; s < KMAX; ++s) cand[base + s] = topi[tid][s];
  }
}

// ------------- kernel 3: per-row bitonic sort of 256 indices + global dup mask -------------
__global__ void __launch_bounds__(256)
fa_sort_kernel(const int* __restrict__ cand, int* __restrict__ sorted,
               uint32_t* __restrict__ colEq) {
  const int i   = blockIdx.x;
  const int tid = threadIdx.x;
  __shared__ int a[256];
  a[tid] = cand[i * 256 + tid];
  __syncthreads();

  for (int kk = 2; kk <= 256; kk <<= 1) {
    for (int jj = kk >> 1; jj > 0; jj >>= 1) {
      const int ixj = tid ^ jj;
      if (ixj > tid) {
        const int x = a[tid], y = a[ixj];
        const bool up = ((tid & kk) == 0);
        if (up ? (x > y) : (x < y)) { a[tid] = y; a[ixj] = x; }
      }
      __syncthreads();
    }
  }

  sorted[i * 256 + tid] = a[tid];
  const int eq = (tid < 255) ? (a[tid + 1] == a[tid]) : 1;   // bit 255 unused, keep 1
  const unsigned mw = (unsigned)__ballot(eq);                // wave32: lanes -> bits
  if ((tid & 31) == 0) atomicAnd(&colEq[tid >> 5], mw);      // AND across all rows
}

// ------------- kernel 4: stable compaction order (kept asc, then dropped asc) -------------
__global__ void fa_order_kernel(const uint32_t* __restrict__ colEq, int* __restrict__ ord) {
  if (threadIdx.x != 0 || blockIdx.x != 0) return;
  int cnt = 0;
  for (int j = 0; j < 256 && cnt < KMAX; ++j) {
    const bool keep = (j == 0) ? true
                               : !((colEq[(j - 1) >> 5] >> ((j - 1) & 31)) & 1u);
    if (keep) ord[cnt++] = j;
  }
  for (int j = 1; j < 256 && cnt < KMAX; ++j) {              // dropped columns, ascending
    const bool keep = !((colEq[(j - 1) >> 5] >> ((j - 1) & 31)) & 1u);
    if (!keep) ord[cnt++] = j;
  }
}

// ------------- kernel 5: emit [1,S,32] -------------
__global__ void fa_emit_kernel(const int* __restrict__ sorted, const int* __restrict__ ord,
                               float* __restrict__ out) {
  const int gid = blockIdx.x * blockDim.x + threadIdx.x;
  if (gid >= S * KMAX) return;
  const int i = gid >> 5, t = gid & 31;
  out[gid] = (float)sorted[i * 256 + ord[t]];
}

extern "C" void kernel_launch(void* const* d_in, const int* in_sizes, int n_in,
                              void* d_out, int out_size, void* d_ws, size_t ws_size,
                              hipStream_t stream) {
  const float* q = (const float*)d_in[0];   // [1,4096,64]
  const float* k = (const float*)d_in[1];   // [1,4096,64]
  const float* W = (const float*)d_in[2];   // [8,8,4]
  (void)d_in[3]; (void)in_sizes; (void)n_in; (void)out_size; (void)ws_size;

  char* ws = (char*)d_ws;
  uint32_t* qhash  = (uint32_t*)(ws + OFF_QHASH);
  uint32_t* khash  = (uint32_t*)(ws + OFF_KHASH);
  uint8_t*  qsign  = (uint8_t*) (ws + OFF_QSIGN);
  uint8_t*  ksign  = (uint8_t*) (ws + OFF_KSIGN);
  uint32_t* colEq  = (uint32_t*)(ws + OFF_COLEQ);
  int*      ord    = (int*)     (ws + OFF_ORD);
  int*      cand   = (int*)     (ws + OFF_CAND);
  int*      sorted = (int*)     (ws + OFF_SORTED);

  fa_hash_kernel<<<dim3((S * NG + 255) / 256, 2), 256, 0, stream>>>(
      q, k, W, qhash, khash, qsign, ksign, colEq);
  fa_cand_kernel<<<dim3(S / TQ, NG), 256, 0, stream>>>(
      q, k, qhash, khash, qsign, ksign, cand);
  fa_sort_kernel<<<S, 256, 0, stream>>>(cand, sorted, colEq);
  fa_order_kernel<<<1, 64, 0, stream>>>(colEq, ord);
  fa_emit_kernel<<<(S * KMAX + 255) / 256, 256, 0, stream>>>(sorted, ord, (float*)d_out);
}